// GNNEncoder_79259326480547
// MI455X (gfx1250) — compile-verified
//
#include <hip/hip_runtime.h>

typedef __attribute__((ext_vector_type(2))) float v2f;
typedef __attribute__((ext_vector_type(8))) float v8f;

// ---------------------------------------------------------------- utilities

__global__ void zero_i32_kernel(int* __restrict__ p, int n) {
    int t = blockIdx.x * blockDim.x + threadIdx.x;
    if (t < n) p[t] = 0;
}

// in-degree count (excluding self loop): cnt[dst]++
__global__ void deg_count_kernel(const int* __restrict__ dst, int* __restrict__ cnt, int E) {
    int t = blockIdx.x * blockDim.x + threadIdx.x;
    if (t < E) atomicAdd(&cnt[dst[t]], 1);
}

// dinv[i] = (deg_with_selfloop)^-1/2 = (cnt[i]+1)^-1/2
__global__ void dinv_kernel(const int* __restrict__ cnt, float* __restrict__ dinv, int n) {
    int t = blockIdx.x * blockDim.x + threadIdx.x;
    if (t < n) dinv[t] = 1.0f / sqrtf((float)(cnt[t] + 1));
}

// single-workgroup exclusive prefix sum: rowptr[0..n] from cnt[0..n-1]
__global__ __launch_bounds__(1024)
void scan_kernel(const int* __restrict__ cnt, int* __restrict__ rowptr, int n) {
    __shared__ int sums[1024];
    const int tid = threadIdx.x;
    const int chunk = (n + 1023) / 1024;
    const int b = tid * chunk;
    const int e = min(b + chunk, n);
    int s = 0;
    for (int i = b; i < e; ++i) s += cnt[i];
    sums[tid] = s;
    __syncthreads();
    for (int off = 1; off < 1024; off <<= 1) {   // Hillis-Steele
        int v = (tid >= off) ? sums[tid - off] : 0;
        __syncthreads();
        sums[tid] += v;
        __syncthreads();
    }
    int run = (tid == 0) ? 0 : sums[tid - 1];
    for (int i = b; i < e; ++i) { rowptr[i] = run; run += cnt[i]; }
    if (tid == 1023) rowptr[n] = run;            // tid 1023's chunk ends at n
}

// fill reverse-CSR with packed metadata: meta[slot] = {src, norm}
__global__ void csr_fill_kernel(const int* __restrict__ src, const int* __restrict__ dst,
                                const float* __restrict__ dinv,
                                const int* __restrict__ rowptr, int* __restrict__ cursor,
                                int2* __restrict__ meta, int E) {
    int t = blockIdx.x * blockDim.x + threadIdx.x;
    if (t >= E) return;
    int s = src[t], d = dst[t];
    int slot = atomicAdd(&cursor[d], 1);
    int2 m;
    m.x = s;
    m.y = __float_as_int(dinv[s] * dinv[d]);
    meta[rowptr[d] + slot] = m;
}

// ------------------------------------------------- WMMA GEMM  h = A * W
// One wave computes a 16-row x full-fout stripe (NCT 16-col tiles): each A
// fragment (global_load_b64) feeds NCT WMMAs, so A is read exactly once.
// Whole W (row-major) lives in LDS; B reads hit consecutive banks per lane.
#define GEMM_WAVES 8
template <int NCT>   // fout = NCT*16
__global__ __launch_bounds__(256)
void gcn_gemm_wmma_kernel(const float* __restrict__ A, const float* __restrict__ W,
                          float* __restrict__ h, int n, int fin) {
    const int fout = NCT * 16;
    __shared__ float Wt[128 * NCT * 16];         // fin<=128; row-major copy of W
    const int tid = threadIdx.x;
    for (int idx = tid; idx < fin * fout; idx += 256)
        Wt[idx] = W[idx];
    __syncthreads();

    const int wv   = tid >> 5;
    const int lane = tid & 31;
    const int tm   = blockIdx.x * GEMM_WAVES + wv;
    const int nrt  = (n + 15) >> 4;
    if (tm >= nrt) return;                       // wave-uniform

    const int half = lane >> 4;                  // 0: K=k,k+1  1: K=k+2,k+3
    const int l    = lane & 15;
    int row  = tm * 16 + l;
    int arow = row < n ? row : n - 1;
    const float* ap = A + (size_t)arow * fin + half * 2;

    v8f acc[NCT];
    #pragma unroll
    for (int t = 0; t < NCT; ++t) acc[t] = {};

    #pragma unroll 2
    for (int k = 0; k < fin; k += 4) {
        v2f a;
        a.x = ap[k];
        a.y = ap[k + 1];
        const int kb = k + half * 2;
        #pragma unroll
        for (int t = 0; t < NCT; ++t) {
            const int col = t * 16 + l;
            v2f b;
            b.x = Wt[kb * fout + col];
            b.y = Wt[(kb + 1) * fout + col];
            acc[t] = __builtin_amdgcn_wmma_f32_16x16x4_f32(
                         false, a, false, b, (short)0, acc[t], false, false);
        }
    }

    const int base_m = tm * 16 + half * 8;
    #pragma unroll
    for (int t = 0; t < NCT; ++t) {
        const int col = t * 16 + l;
        #pragma unroll
        for (int j = 0; j < 8; ++j) {
            int m = base_m + j;
            if (m < n) h[(size_t)m * fout + col] = acc[t][j];
        }
    }
}

// ----------------------- fused aggregate: out = [relu](selfloop + gather + b)
// One wave per node; lane owns CPL contiguous channels. Edge metadata is a
// single wave-uniform b64 load; feature gathers are contiguous b128/b64.
template <int CPL>
__global__ void gcn_gather_kernel(const float* __restrict__ h,
                                  const int* __restrict__ rowptr,
                                  const int2* __restrict__ meta,
                                  const float* __restrict__ dinv,
                                  const float* __restrict__ bias,
                                  float* __restrict__ out,
                                  int n, int fout, int do_relu) {
    const int wave = (blockIdx.x * blockDim.x + threadIdx.x) >> 5;
    const int lane = threadIdx.x & 31;
    if (wave >= n) return;                       // wave-uniform
    const int i  = wave;
    const int c0 = lane * CPL;

    float acc[CPL];
    const float dv = dinv[i];
    const float* hi = h + (size_t)i * fout + c0;
    #pragma unroll
    for (int q = 0; q < CPL; ++q) acc[q] = hi[q] * dv * dv;   // self loop (dinv^2)

    const int beg = rowptr[i], end = rowptr[i + 1];
    #pragma unroll 2
    for (int j = beg; j < end; ++j) {
        const int2 m  = meta[j];
        const float nm = __int_as_float(m.y);
        const float* hs = h + (size_t)m.x * fout + c0;
        #pragma unroll
        for (int q = 0; q < CPL; ++q) acc[q] = fmaf(hs[q], nm, acc[q]);
    }

    float* oi = out + (size_t)i * fout + c0;
    #pragma unroll
    for (int q = 0; q < CPL; ++q) {
        float v = acc[q] + bias[c0 + q];
        oi[q] = do_relu ? fmaxf(v, 0.0f) : v;
    }
}

// ---------------------------------------------------------------- launcher

static inline int cdiv(long long a, int b) { return (int)((a + b - 1) / b); }
static inline size_t align4up(size_t x) { return (x + 3) & ~(size_t)3; }

extern "C" void kernel_launch(void* const* d_in, const int* in_sizes, int n_in,
                              void* d_out, int out_size, void* d_ws, size_t ws_size,
                              hipStream_t stream) {
    const float* x  = (const float*)d_in[0];
    const int*   ei = (const int*)  d_in[1];
    const float* W1 = (const float*)d_in[2];
    const float* b1 = (const float*)d_in[3];
    const float* W2 = (const float*)d_in[4];
    const float* b2 = (const float*)d_in[5];
    const float* W3 = (const float*)d_in[6];
    const float* b3 = (const float*)d_in[7];

    const int IN_C = 128, HID = 128, MID = 64, OUT = 64;
    const int n = in_sizes[0] / IN_C;            // 100000
    const int E = in_sizes[1] / 2;               // 1000000
    const int* src = ei;
    const int* dst = ei + E;

    // workspace carve-up (4B units)
    char* base = (char*)d_ws;
    size_t off = 0;
    auto take = [&](size_t elems) { void* p = base + off * 4; off += align4up(elems); return p; };
    int*   cnt    = (int*)  take(n);             // degree count, then CSR cursor
    int*   rowptr = (int*)  take(n + 1);
    float* dinv   = (float*)take(n);
    int2*  meta   = (int2*) take((size_t)2 * E); // packed {src, norm}
    float* h      = (float*)take((size_t)n * 128);
    float* xn     = (float*)take((size_t)n * 128);
    (void)ws_size;

    const int B = 256;

    // ---- build normalization + reverse CSR (once; reused by all 3 layers)
    zero_i32_kernel<<<cdiv(n, B), B, 0, stream>>>(cnt, n);
    deg_count_kernel<<<cdiv(E, B), B, 0, stream>>>(dst, cnt, E);
    dinv_kernel<<<cdiv(n, B), B, 0, stream>>>(cnt, dinv, n);
    scan_kernel<<<1, 1024, 0, stream>>>(cnt, rowptr, n);
    zero_i32_kernel<<<cdiv(n, B), B, 0, stream>>>(cnt, n);          // reuse as cursor
    csr_fill_kernel<<<cdiv(E, B), B, 0, stream>>>(src, dst, dinv, rowptr, cnt, meta, E);

    const int nrt = (n + 15) / 16;
    const int gemm_blocks = cdiv(nrt, GEMM_WAVES);
    const int gather_blocks = cdiv((long long)n * 32, B);

    auto run_layer = [&](const float* in, const float* W, const float* bias,
                         float* out, int fin, int fout, int relu) {
        if (fout == 128)
            gcn_gemm_wmma_kernel<8><<<gemm_blocks, B, 0, stream>>>(in, W, h, n, fin);
        else
            gcn_gemm_wmma_kernel<4><<<gemm_blocks, B, 0, stream>>>(in, W, h, n, fin);
        if (fout == 128)
            gcn_gather_kernel<4><<<gather_blocks, B, 0, stream>>>(h, rowptr, meta,
                                                                  dinv, bias, out, n, fout, relu);
        else
            gcn_gather_kernel<2><<<gather_blocks, B, 0, stream>>>(h, rowptr, meta,
                                                                  dinv, bias, out, n, fout, relu);
    };

    run_layer(x,  W1, b1, xn,            IN_C, HID, 1);   // 128 -> 128, relu
    run_layer(xn, W2, b2, xn,            HID,  MID, 1);   // 128 ->  64, relu
    run_layer(xn, W3, b3, (float*)d_out, MID,  OUT, 0);   //  64 ->  64
}